// Conv_SNU_Network_classification_16561393893856
// MI455X (gfx1250) — compile-verified
//
#include <hip/hip_runtime.h>
#include <hip/hip_bf16.h>
#include <cmath>

typedef float v2f __attribute__((ext_vector_type(2)));
typedef float v8f __attribute__((ext_vector_type(8)));

#define LTAU    0.8f
#define TSTEPS  20
#define CONV_HW 23
#define CONV_N  529      // 23*23
#define NTILES  34       // ceil(529/16)
#define CCH     6
#define POOLW   11
#define FEATN   726      // 6*11*11
#define NOUT    2
#define BATCH   1024
#define XPAD    (33 * 32)   // frame + 1 extra zero row for padded-column reads

__device__ __forceinline__ float sigm(float v) {
    return 1.0f / (1.0f + __expf(-v));
}

__global__ __launch_bounds__(256)
void snu_forward_kernel(const float* __restrict__ x,       // [B,1024,T]
                        const float* __restrict__ cn1_w,   // [6,1,10,10]
                        const float* __restrict__ cn1_b,   // [6]
                        const float* __restrict__ l2_w,    // [2,726]
                        const float* __restrict__ l2_b,    // [2]
                        float* __restrict__ m_out,         // [B,2]
                        float* __restrict__ out_rec)       // [B,21,2]
{
    __shared__ float xlds[XPAD];               // current frame (+zero pad row)
    __shared__ float Wlds[16 * 100];           // conv weights, M padded 6->16
    __shared__ float W2lds[NOUT * FEATN];      // dense weights
    __shared__ float s1[CCH * CONV_N];         // membrane state, conv layer
    __shared__ float y1[CCH * CONV_N];         // spike state, conv layer
    __shared__ float hbuf[FEATN];              // pooled features
    __shared__ float b1s[CCH];
    __shared__ float b2s[NOUT];
    __shared__ float s2s[NOUT];
    __shared__ float y2s[NOUT];
    __shared__ float maccs[NOUT];

    const int bimg = blockIdx.x;
    const int tid  = threadIdx.x;
    const int wave = tid >> 5;
    const int lane = tid & 31;
    const int hi   = (lane >= 16) ? 1 : 0;     // high half-wave flag

    // ---- one-time LDS init -------------------------------------------------
    for (int i = tid; i < 16 * 100; i += 256) {
        int mrow = i / 100, k = i - mrow * 100;
        Wlds[i] = (mrow < CCH) ? cn1_w[mrow * 100 + k] : 0.0f;
    }
    for (int i = 1024 + tid; i < XPAD; i += 256) xlds[i] = 0.0f;  // pad row
    for (int i = tid; i < NOUT * FEATN; i += 256) W2lds[i] = l2_w[i];
    for (int i = tid; i < CCH * CONV_N; i += 256) { s1[i] = 0.0f; y1[i] = 0.0f; }
    if (tid < CCH) b1s[tid] = cn1_b[tid];
    if (tid < NOUT) {
        b2s[tid] = l2_b[tid];
        s2s[tid] = 0.0f; y2s[tid] = 0.0f; maccs[tid] = 0.0f;
        out_rec[((size_t)bimg * (TSTEPS + 1)) * NOUT + tid] = 0.0f;  // t = 0 row
    }
    __syncthreads();

    // ---- A fragments: loaded ONCE, live across the whole time loop ---------
    // K permutation: chunk kk = (r, c2), r,c2 in 0..4 covers the 2x2 tap block
    //   low  half-wave (lanes 0-15):  taps (2r,   2c2), (2r,   2c2+1)
    //   high half-wave (lanes 16-31): taps (2r+1, 2c2), (2r+1, 2c2+1)
    // This matches the 16x4 f32 A layout {K0,K1 | K2,K3} and makes every LDS
    // offset a compile-time immediate after a single per-lane base adjust.
    const int mA = lane & 15;
    v2f afrag[25];
    {
        const float* Wp = Wlds + mA * 100 + hi * 10;
#pragma unroll
        for (int r = 0; r < 5; ++r)
#pragma unroll
            for (int c2 = 0; c2 < 5; ++c2) {
                afrag[r * 5 + c2].x = Wp[20 * r + 2 * c2];
                afrag[r * 5 + c2].y = Wp[20 * r + 2 * c2 + 1];
            }
    }
    __syncthreads();

    const float* xb = x + (size_t)bimg * 1024 * TSTEPS;

    for (int t = 0; t < TSTEPS; ++t) {
        // ---- load frame x[:, :, t] (stride T over pixels) ------------------
        for (int p = tid; p < 1024; p += 256) {
            xlds[p] = xb[(size_t)p * TSTEPS + t];
            if (t + 1 < TSTEPS)
                __builtin_prefetch(xb + (size_t)p * TSTEPS + t + 1, 0, 0);
        }
        __syncthreads();

        // ---- conv as GEMM [16 x 100] x [100 x 529] via V_WMMA_F32_16X16X4 --
        for (int tile = wave; tile < NTILES; tile += 8) {
            const int n  = tile * 16 + (lane & 15);     // padded cols read zeros
            const int oi = n / CONV_HW;
            const int oj = n - oi * CONV_HW;
            const float* bp = xlds + (oi + hi) * 32 + oj;

            v8f acc = {0.f, 0.f, 0.f, 0.f, 0.f, 0.f, 0.f, 0.f};
#pragma unroll
            for (int r = 0; r < 5; ++r) {
                v2f bfr[5];
#pragma unroll
                for (int c2 = 0; c2 < 5; ++c2) {
                    bfr[c2].x = bp[64 * r + 2 * c2];
                    bfr[c2].y = bp[64 * r + 2 * c2 + 1];
                }
#pragma unroll
                for (int c2 = 0; c2 < 5; ++c2)
                    acc = __builtin_amdgcn_wmma_f32_16x16x4_f32(
                        false, afrag[r * 5 + c2], false, bfr[c2],
                        (short)0, acc, false, false);
            }

            // fused SNU membrane update (rows 0..5 live in lanes 0-15)
            if (lane < 16 && n < CONV_N) {
#pragma unroll
                for (int c = 0; c < CCH; ++c) {
                    const int idx = c * CONV_N + n;
                    float sp = s1[idx], yp = y1[idx];
                    float s  = acc[c] + LTAU * sp * (1.0f - yp);
                    s = fmaxf(s, 0.0f);
                    s1[idx] = s;
                    y1[idx] = sigm(s + b1s[c]);
                }
            }
        }
        __syncthreads();

        // ---- 2x2 max pool (VALID, stride 2) -> hbuf ------------------------
        for (int f = tid; f < FEATN; f += 256) {
            int c   = f / 121;
            int rem = f - c * 121;
            int i   = rem / POOLW;
            int j   = rem - i * POOLW;
            const float* base = y1 + c * CONV_N + (2 * i) * CONV_HW + 2 * j;
            hbuf[f] = fmaxf(fmaxf(base[0], base[1]),
                            fmaxf(base[CONV_HW], base[CONV_HW + 1]));
        }
        __syncthreads();

        // ---- dense SNU (2 outputs), one wave per output --------------------
        if (tid < 64) {
            const int o = tid >> 5;
            float p = 0.0f;
            for (int f = lane; f < FEATN; f += 32)
                p += hbuf[f] * W2lds[o * FEATN + f];
#pragma unroll
            for (int off = 16; off > 0; off >>= 1)
                p += __shfl_down(p, off, 32);
            if (lane == 0) {
                float sp = s2s[o], yp = y2s[o];
                float s  = p + LTAU * sp * (1.0f - yp);
                s = fmaxf(s, 0.0f);
                float yv = sigm(s + b2s[o]);
                s2s[o] = s; y2s[o] = yv;
                maccs[o] += yv;
                out_rec[((size_t)bimg * (TSTEPS + 1) + (t + 1)) * NOUT + o] = yv;
            }
        }
        __syncthreads();
    }

    if (tid < NOUT)
        m_out[bimg * NOUT + tid] = maccs[tid] * (1.0f / TSTEPS);
}

// ---- batch-wide loss / accuracy reduction ----------------------------------
__global__ __launch_bounds__(256)
void snu_loss_kernel(const float* __restrict__ m,   // [B,2]
                     const int* __restrict__ y,     // [B]
                     float* __restrict__ loss_out,
                     float* __restrict__ acc_out)
{
    __shared__ float lsum[256];
    __shared__ float asum[256];
    const int tid = threadIdx.x;
    float l = 0.0f, a = 0.0f;
    for (int b = tid; b < BATCH; b += 256) {
        float m0 = m[b * 2 + 0], m1 = m[b * 2 + 1];
        int lab = y[b];
        float mx  = fmaxf(m0, m1);
        float lse = mx + logf(__expf(m0 - mx) + __expf(m1 - mx));
        float my  = lab ? m1 : m0;
        l += -(my - lse);
        int pred = (m1 > m0) ? 1 : 0;   // argmax with first-max tie-break
        a += (pred == lab) ? 1.0f : 0.0f;
    }
    lsum[tid] = l; asum[tid] = a;
    __syncthreads();
    for (int s = 128; s > 0; s >>= 1) {
        if (tid < s) { lsum[tid] += lsum[tid + s]; asum[tid] += asum[tid + s]; }
        __syncthreads();
    }
    if (tid == 0) {
        *loss_out = lsum[0] * (1.0f / BATCH);
        *acc_out  = asum[0] * (1.0f / BATCH);
    }
}

extern "C" void kernel_launch(void* const* d_in, const int* in_sizes, int n_in,
                              void* d_out, int out_size, void* d_ws, size_t ws_size,
                              hipStream_t stream) {
    const float* x     = (const float*)d_in[0];
    const int*   y     = (const int*)d_in[1];
    const float* cn1_w = (const float*)d_in[2];
    const float* cn1_b = (const float*)d_in[3];
    const float* l2_w  = (const float*)d_in[4];
    const float* l2_b  = (const float*)d_in[5];

    // output layout (flat, return order): loss[1], m[1024*2], out_rec[1024*21*2], acc[1]
    float* out      = (float*)d_out;
    float* loss     = out;
    float* m        = out + 1;
    float* out_rec  = out + 1 + BATCH * NOUT;
    float* acc      = out + 1 + BATCH * NOUT + BATCH * (TSTEPS + 1) * NOUT;

    snu_forward_kernel<<<BATCH, 256, 0, stream>>>(x, cn1_w, cn1_b, l2_w, l2_b,
                                                  m, out_rec);
    snu_loss_kernel<<<1, 256, 0, stream>>>(m, y, loss, acc);
}